// GNN_18786186952946
// MI455X (gfx1250) — compile-verified
//
#include <hip/hip_runtime.h>
#include <hip/hip_bf16.h>

typedef __attribute__((ext_vector_type(2))) float v2f;
typedef __attribute__((ext_vector_type(8))) float v8f;

#define CH 128       // hidden/in channels
#define LDSK 132     // padded LDS row stride (conflict-free ds_load_b64)

// ---------------------------------------------------------------------------
// Generic fp32 WMMA GEMM: D[M,N] = A[M,K] * B[K,N] (+bias) (+relu)
// Block: dim3(32, N/16)  -> one wave per 16-col tile, block covers 16 rows x N cols
// Grid:  dim3(1, M/16)
// M, N multiples of 16; K multiple of 4; K <= 128.
// Uses V_WMMA_F32_16X16X4_F32, K-loop in steps of 4.
// ---------------------------------------------------------------------------
__global__ void gcn_wmma_gemm(const float* __restrict__ A, const float* __restrict__ B,
                              const float* __restrict__ bias, float* __restrict__ D,
                              int M, int N, int K, int fuse_relu) {
    __shared__ float aTile[16 * LDSK];

    const int tm = blockIdx.y * 16;
    const int nthreads = blockDim.x * blockDim.y;
    const int tid = threadIdx.y * 32 + threadIdx.x;

    // cooperative, coalesced load of the 16xK A tile into LDS (padded rows)
    for (int i = tid; i < 16 * K; i += nthreads) {
        int r = i / K;
        int k = i - r * K;
        aTile[r * LDSK + k] = A[(size_t)(tm + r) * K + k];
    }
    __syncthreads();

    const int lane  = threadIdx.x;            // 0..31
    const int tn    = threadIdx.y * 16;       // this wave's column tile
    const int col   = tn + (lane & 15);
    const int khalf = (lane >> 4) << 1;       // lanes 16-31 handle K=+2,+3
    const int arow  = lane & 15;

    v8f acc = {};
    for (int k0 = 0; k0 < K; k0 += 4) {
        const int ka = k0 + khalf;
        // A: 16x4 fp32 tile; lane holds A[arow][ka], A[arow][ka+1]
        float2 av = *(const float2*)&aTile[arow * LDSK + ka];
        v2f a; a.x = av.x; a.y = av.y;
        // B: 4x16 fp32 tile; lane holds B[ka][col], B[ka+1][col]
        v2f b;
        b.x = B[(size_t)ka * N + col];
        b.y = B[(size_t)(ka + 1) * N + col];
        acc = __builtin_amdgcn_wmma_f32_16x16x4_f32(
            /*neg_a=*/false, a, /*neg_b=*/false, b,
            /*c_mod=*/(short)0, acc, /*reuse_a=*/false, /*reuse_b=*/false);
    }

    const float bb = bias ? bias[col] : 0.0f;
    const int rbase = tm + ((lane >> 4) << 3);   // lanes 16-31 -> rows M=8..15
#pragma unroll
    for (int v = 0; v < 8; ++v) {
        float val = acc[v] + bb;
        if (fuse_relu) val = fmaxf(val, 0.0f);
        D[(size_t)(rbase + v) * N + col] = val;
    }
}

// ---------------------------------------------------------------------------
// Degree / normalization
// ---------------------------------------------------------------------------
__global__ void gcn_count_deg(const int* __restrict__ dst, float* __restrict__ deg, int E) {
    int e = blockIdx.x * blockDim.x + threadIdx.x;
    if (e < E) atomicAdd(&deg[dst[e]], 1.0f);
}

__global__ void gcn_make_dinv(float* __restrict__ deg, int N) {
    int n = blockIdx.x * blockDim.x + threadIdx.x;
    if (n < N) deg[n] = rsqrtf(deg[n] + 1.0f);   // +1 = self-loop; always > 0
}

// ---------------------------------------------------------------------------
// Edge scatter: one wave per edge, lane handles 4 channels (float4).
// out[dst] += h[src] * dinv[src]*dinv[dst]
// ---------------------------------------------------------------------------
__global__ void gcn_edge_scatter(const int* __restrict__ src, const int* __restrict__ dst,
                                 const float* __restrict__ dinv,
                                 const float* __restrict__ h, float* __restrict__ out, int E) {
    int t = blockIdx.x * blockDim.x + threadIdx.x;
    int e = t >> 5;
    if (e >= E) return;
    int lane = t & 31;
    int s = src[e];
    int d = dst[e];
    float coef = dinv[s] * dinv[d];
    float4 v = ((const float4*)(h + (size_t)s * CH))[lane];
    float* o = out + (size_t)d * CH + lane * 4;
    atomicAdd(o + 0, v.x * coef);
    atomicAdd(o + 1, v.y * coef);
    atomicAdd(o + 2, v.z * coef);
    atomicAdd(o + 3, v.w * coef);
}

// ---------------------------------------------------------------------------
// finalize: out = relu(agg + tmp*dinv^2 + bias)  (self-loop term folded in)
// ---------------------------------------------------------------------------
__global__ void gcn_finalize(const float* __restrict__ agg, const float* __restrict__ tmp,
                             const float* __restrict__ dinv, const float* __restrict__ bias,
                             float* __restrict__ out, int total) {
    int i = blockIdx.x * blockDim.x + threadIdx.x;
    if (i >= total) return;
    int n = i >> 7;        // / CH
    int c = i & (CH - 1);
    float di = dinv[n];
    float v = agg[i] + tmp[i] * di * di + bias[c];
    out[i] = fmaxf(v, 0.0f);
}

// ---------------------------------------------------------------------------
// Global mean pool: scatter-sum per graph + count, then divide.
// ---------------------------------------------------------------------------
__global__ void gcn_pool_sum(const float* __restrict__ h, const int* __restrict__ batch,
                             float* __restrict__ pooled, float* __restrict__ cnt, int N) {
    int t = blockIdx.x * blockDim.x + threadIdx.x;
    int n = t >> 5;
    if (n >= N) return;
    int lane = t & 31;
    int g = batch[n];
    float4 v = ((const float4*)(h + (size_t)n * CH))[lane];
    float* p = pooled + (size_t)g * CH + lane * 4;
    atomicAdd(p + 0, v.x);
    atomicAdd(p + 1, v.y);
    atomicAdd(p + 2, v.z);
    atomicAdd(p + 3, v.w);
    if (lane == 0) atomicAdd(&cnt[g], 1.0f);
}

__global__ void gcn_pool_div(float* __restrict__ pooled, const float* __restrict__ cnt, int total) {
    int i = blockIdx.x * blockDim.x + threadIdx.x;
    if (i >= total) return;
    pooled[i] /= fmaxf(cnt[i >> 7], 1.0f);
}

// ---------------------------------------------------------------------------
// Host launcher
// ---------------------------------------------------------------------------
extern "C" void kernel_launch(void* const* d_in, const int* in_sizes, int n_in,
                              void* d_out, int out_size, void* d_ws, size_t ws_size,
                              hipStream_t stream) {
    const float* x    = (const float*)d_in[0];
    const float* W1   = (const float*)d_in[1];
    const float* b1   = (const float*)d_in[2];
    const float* W2   = (const float*)d_in[3];
    const float* b2   = (const float*)d_in[4];
    const float* Wlin = (const float*)d_in[5];
    const float* blin = (const float*)d_in[6];
    const int*   ei   = (const int*)d_in[7];
    const int*   batch = (const int*)d_in[8];

    const int OUTC = 64;
    const int G    = 512;
    const int N    = in_sizes[0] / CH;   // 50000
    const int E    = in_sizes[7] / 2;    // 600000
    const int* src = ei;
    const int* dst = ei + E;

    // workspace layout (floats)
    float* tmp    = (float*)d_ws;                       // [N*CH]  h_in @ W
    float* agg    = tmp + (size_t)N * CH;               // [N*CH]  scatter target
    float* h      = agg + (size_t)N * CH;               // [N*CH]  layer output
    float* dinv   = h + (size_t)N * CH;                 // [N]
    float* pooled = dinv + N;                           // [G*CH]
    float* cnt    = pooled + (size_t)G * CH;            // [G]

    const int nodeElems = N * CH;

    // --- degree normalization -------------------------------------------
    hipMemsetAsync(dinv, 0, sizeof(float) * (size_t)N, stream);
    gcn_count_deg<<<(E + 255) / 256, 256, 0, stream>>>(dst, dinv, E);
    gcn_make_dinv<<<(N + 255) / 256, 256, 0, stream>>>(dinv, N);

    const dim3 blkGemm(32, CH / 16);          // 8 waves, 256 threads
    const dim3 grdGemm(1, N / 16);            // 3125 row tiles

    // --- layer 1 ---------------------------------------------------------
    gcn_wmma_gemm<<<grdGemm, blkGemm, 0, stream>>>(x, W1, nullptr, tmp, N, CH, CH, 0);
    hipMemsetAsync(agg, 0, sizeof(float) * (size_t)nodeElems, stream);
    gcn_edge_scatter<<<(E * 32 + 255) / 256, 256, 0, stream>>>(src, dst, dinv, tmp, agg, E);
    gcn_finalize<<<(nodeElems + 255) / 256, 256, 0, stream>>>(agg, tmp, dinv, b1, h, nodeElems);

    // --- layer 2 ---------------------------------------------------------
    gcn_wmma_gemm<<<grdGemm, blkGemm, 0, stream>>>(h, W2, nullptr, tmp, N, CH, CH, 0);
    hipMemsetAsync(agg, 0, sizeof(float) * (size_t)nodeElems, stream);
    gcn_edge_scatter<<<(E * 32 + 255) / 256, 256, 0, stream>>>(src, dst, dinv, tmp, agg, E);
    gcn_finalize<<<(nodeElems + 255) / 256, 256, 0, stream>>>(agg, tmp, dinv, b2, h, nodeElems);

    // --- global mean pool ------------------------------------------------
    hipMemsetAsync(pooled, 0, sizeof(float) * ((size_t)G * CH + G), stream);
    gcn_pool_sum<<<(N * 32 + 255) / 256, 256, 0, stream>>>(h, batch, pooled, cnt, N);
    gcn_pool_div<<<(G * CH + 255) / 256, 256, 0, stream>>>(pooled, cnt, G * CH);

    // --- head: [512,128] @ [128,64] + blin -> d_out ----------------------
    gcn_wmma_gemm<<<dim3(1, G / 16), dim3(32, OUTC / 16), 0, stream>>>(
        pooled, Wlin, blin, (float*)d_out, G, OUTC, CH, 0);
}